// SelectiveSSM_32530082300491
// MI455X (gfx1250) — compile-verified
//
#include <hip/hip_runtime.h>
#include <hip/hip_bf16.h>
#include <math.h>

typedef float v2f __attribute__((ext_vector_type(2)));
typedef float v8f __attribute__((ext_vector_type(8)));
typedef int   v4i __attribute__((ext_vector_type(4)));

#define B_   8
#define L_   4096
#define D_   1024
#define N_   16
#define K_   4
#define CIN  80       // 2*INNER + N
#define DCHUNK 256
#define LDSTR  260    // 256 + 4 pad -> (m*260)%64 = 4m, conflict-free A-frag reads
#define XROWS  19     // rows l0-1 .. l0+17 staged for the 4-tap conv window

#define AS1 __attribute__((address_space(1)))
#define AS3 __attribute__((address_space(3)))

#if __has_builtin(__builtin_amdgcn_global_load_async_to_lds_b128)
#define HAVE_ASYNC_LDS 1
#endif

// ---------------------------------------------------------------------------
// Kernel A: depthwise conv (x staged once via async global->LDS DMA) + input
// projection GEMM (32768 x 1024) x (1024 x 80) via v_wmma_f32_16x16x4_f32,
// + pointwise SSM parameter computation. Outputs A_bar / B_bar / Cmean in
// (B, N, L) layout so the scan kernel reads contiguously.
// ---------------------------------------------------------------------------
__global__ __launch_bounds__(160)
void ssm_conv_proj_kernel(const float* __restrict__ x,
                          const float* __restrict__ conv_w,   // (D,1,K)
                          const float* __restrict__ conv_b,   // (D)
                          const float* __restrict__ Win,      // (D, 80)
                          const float* __restrict__ b_in,     // (80)
                          const float* __restrict__ A_log,    // (N)
                          float* __restrict__ abar,           // (B,N,L)
                          float* __restrict__ bbar,           // (B,N,L)
                          float* __restrict__ cmean)          // (B,N,L)
{
    __shared__ float xraw[XROWS][DCHUNK];  // raw x window (rows l0-1..l0+17)
    __shared__ float xcs[16][LDSTR];       // conv-output chunk (16 rows x 256 d)
    __shared__ float projs[16][CIN];       // staged proj tile for epilogue

    const int tile = blockIdx.x;         // 2048 tiles over B*L rows
    const int b    = tile >> 8;          // 256 tiles per batch (L/16)
    const int l0   = (tile & 255) << 4;
    const int tid  = threadIdx.x;
    const int lane = tid & 31;
    const int wave = tid >> 5;           // 0..4, one 16-col strip each
    const int m    = lane & 15;          // row within tile (A frag / B col)
    const int kb   = (lane >> 4) << 1;   // K sub-offset: 0 or 2 (ISA 16x4 layout)
    const int col  = wave * 16 + m;      // this lane's output column

    v8f acc = {};

    for (int dbase = 0; dbase < D_; dbase += DCHUNK) {
        // --- stage raw x window into LDS: one async B128 per 4 floats.
        // Boundary rows (conv zero-pad) are zero-filled with plain LDS stores.
        for (int idx = tid; idx < XROWS * (DCHUNK / 4); idx += 160) {
            int row = idx / (DCHUNK / 4);
            int seg = (idx - row * (DCHUNK / 4)) << 2;   // float offset, 16B aligned
            int li  = l0 + row - 1;
            if (li >= 0 && li < L_) {
                const float* gsrc = x + ((size_t)(b * L_ + li)) * D_ + dbase + seg;
#ifdef HAVE_ASYNC_LDS
                __builtin_amdgcn_global_load_async_to_lds_b128(
                    (AS1 v4i*)gsrc, (AS3 v4i*)&xraw[row][seg], 0, 0);
#else
                xraw[row][seg + 0] = gsrc[0];
                xraw[row][seg + 1] = gsrc[1];
                xraw[row][seg + 2] = gsrc[2];
                xraw[row][seg + 3] = gsrc[3];
#endif
            } else {
                xraw[row][seg + 0] = 0.f;
                xraw[row][seg + 1] = 0.f;
                xraw[row][seg + 2] = 0.f;
                xraw[row][seg + 3] = 0.f;
            }
        }
#ifdef HAVE_ASYNC_LDS
#if __has_builtin(__builtin_amdgcn_s_wait_asynccnt)
        __builtin_amdgcn_s_wait_asynccnt(0);
#else
        asm volatile("s_wait_asynccnt 0x0" ::: "memory");
#endif
#endif
        __syncthreads();

        // --- depthwise conv from LDS, rolling 4-tap window per column
        for (int dc = tid; dc < DCHUNK; dc += 160) {
            int d = dbase + dc;
            float w0 = conv_w[d * K_ + 0];
            float w1 = conv_w[d * K_ + 1];
            float w2 = conv_w[d * K_ + 2];
            float w3 = conv_w[d * K_ + 3];
            float cb = conv_b[d];
            float xm1 = xraw[0][dc];
            float x0  = xraw[1][dc];
            float x1  = xraw[2][dc];
#pragma unroll
            for (int r = 0; r < 16; ++r) {
                float x2 = xraw[r + 3][dc];
                float v = xm1 * w0 + x0 * w1 + x1 * w2 + x2 * w3 + cb;
                if (l0 + r == L_ - 1) v = 0.f;   // Lc = L-1: last row is jnp.pad zero
                xcs[r][dc] = v;
                xm1 = x0; x0 = x1; x1 = x2;
            }
        }
        __syncthreads();

        // --- GEMM: (16 x 256 chunk) x (256 x 16 strip of Win), f32 WMMA
        for (int kk = 0; kk < DCHUNK; kk += 4) {
            v2f a, bf;
            a.x = xcs[m][kk + kb];
            a.y = xcs[m][kk + kb + 1];
            const float* wp = Win + (size_t)(dbase + kk + kb) * CIN + col;
            bf.x = wp[0];
            bf.y = wp[CIN];
            acc = __builtin_amdgcn_wmma_f32_16x16x4_f32(
                false, a, false, bf, (short)0, acc, false, false);
        }
        __syncthreads();
    }

    // --- stage proj(+bias) tile (C layout: VGPR v -> rows v / v+8)
    {
        float bias = b_in[col];
#pragma unroll
        for (int vv = 0; vv < 8; ++vv) {
            int mr = vv + ((lane >> 4) << 3);
            projs[mr][col] = acc[vv] + bias;
        }
    }
    __syncthreads();

    // --- pointwise SSM parameters
    for (int idx = tid; idx < 16 * N_; idx += 160) {
        int r = idx >> 4, n = idx & 15;
        float delta = projs[r][n];
        float Bv = 0.5f * (projs[r][N_ + 2 * n]      + projs[r][N_ + 2 * n + 1]);
        float Cv = 0.5f * (projs[r][N_ + 32 + 2 * n] + projs[r][N_ + 32 + 2 * n + 1]);
        float A  = -__expf(A_log[n]);
        float sp = (delta > 20.f) ? delta : log1pf(__expf(delta));  // softplus
        float dA = sp * A;
        float Ab = __expf(dA);
        const float eps = 1e-4f;
        float sgn   = (A > 0.f) ? 1.f : ((A < 0.f) ? -1.f : 0.f);
        float Asafe = (fabsf(A) < eps) ? (eps * sgn + eps) : A;
        float taylor = 1.f + 0.5f * dA + dA * dA * (1.f / 6.f);
        float scale  = (fabsf(A) < eps) ? taylor : (Ab - 1.f) / Asafe;
        size_t o = ((size_t)(b * N_ + n)) * L_ + (l0 + r);
        abar[o]  = Ab;
        bbar[o]  = scale * Bv;
        cmean[o] = Cv;
    }
}

// ---------------------------------------------------------------------------
// Kernel B: linear-recurrence scan h[t] = A_bar[t]*h[t-1] + B_bar[t] along L.
// One wave32 per (b, n): 32 lanes x 128-step chunks; affine-map composition
// scanned across lanes with __shfl_up (wave32). Writes C*h as (B, L, N).
// ---------------------------------------------------------------------------
__global__ __launch_bounds__(32)
void ssm_scan_kernel(const float* __restrict__ abar,
                     const float* __restrict__ bbar,
                     const float* __restrict__ cmean,
                     float* __restrict__ ch)            // (B,L,N)
{
    const int bn   = blockIdx.x;     // 0..127 = b*16+n
    const int b    = bn >> 4;
    const int n    = bn & 15;
    const int lane = threadIdx.x;
    const size_t base = ((size_t)bn) * L_;
    const int CH   = L_ / 32;        // 128
    const int lbeg = lane * CH;

    // pass 1: compose this chunk's affine map (h_out = a*h_in + hb)
    float a = 1.f, hb = 0.f;
    for (int t = 0; t < CH; ++t) {
        float A  = abar[base + lbeg + t];
        float Bv = bbar[base + lbeg + t];
        hb = A * hb + Bv;
        a  = A * a;
    }
    // pass 2: inclusive scan of affine maps across the wave
    float ai = a, bi = hb;
    for (int off = 1; off < 32; off <<= 1) {
        float ap = __shfl_up(ai, off, 32);
        float bp = __shfl_up(bi, off, 32);
        if (lane >= off) { bi = ai * bp + bi; ai = ai * ap; }
    }
    float hin = __shfl_up(bi, 1, 32);   // exclusive prefix: h entering this chunk
    if (lane == 0) hin = 0.f;

    // pass 3: replay chunk with true carry-in; emit C*h
    float h = hin;
    for (int t = 0; t < CH; ++t) {
        size_t o = base + lbeg + t;
        h = abar[o] * h + bbar[o];
        int l = lbeg + t;
        ch[((size_t)(b * L_ + l)) * N_ + n] = cmean[o] * h;
    }
}

// ---------------------------------------------------------------------------
// Kernel C: output projection (32768 x 16) x (16 x 1024) + b_out via f32 WMMA.
// 8 waves per block, each owning one 16-col N-tile; ch tile stays in L2.
// ---------------------------------------------------------------------------
__global__ __launch_bounds__(256)
void ssm_out_kernel(const float* __restrict__ ch,     // (B*L, 16)
                    const float* __restrict__ Wout,   // (16, 1024)
                    const float* __restrict__ b_out,  // (1024)
                    float* __restrict__ y)            // (B*L, 1024)
{
    const int row0 = blockIdx.x << 4;                 // over B*L
    const int tid = threadIdx.x;
    const int lane = tid & 31;
    const int wave = tid >> 5;
    const int col0 = (blockIdx.y * 8 + wave) << 4;
    const int m  = lane & 15;
    const int kb = (lane >> 4) << 1;
    const int col = col0 + m;

    v8f acc = {};
#pragma unroll
    for (int k4 = 0; k4 < N_; k4 += 4) {
        v2f a, bf;
        const float* ap = ch + (size_t)(row0 + m) * N_ + k4 + kb;
        a.x = ap[0];
        a.y = ap[1];
        const float* wp = Wout + (size_t)(k4 + kb) * D_ + col;
        bf.x = wp[0];
        bf.y = wp[D_];
        acc = __builtin_amdgcn_wmma_f32_16x16x4_f32(
            false, a, false, bf, (short)0, acc, false, false);
    }
    float bias = b_out[col];
#pragma unroll
    for (int vv = 0; vv < 8; ++vv) {
        int mr = vv + ((lane >> 4) << 3);
        y[(size_t)(row0 + mr) * D_ + col] = acc[vv] + bias;
    }
}

// ---------------------------------------------------------------------------
extern "C" void kernel_launch(void* const* d_in, const int* in_sizes, int n_in,
                              void* d_out, int out_size, void* d_ws, size_t ws_size,
                              hipStream_t stream) {
    const float* x      = (const float*)d_in[0];
    const float* conv_w = (const float*)d_in[1];
    const float* conv_b = (const float*)d_in[2];
    const float* Win    = (const float*)d_in[3];
    const float* b_in   = (const float*)d_in[4];
    const float* Wout   = (const float*)d_in[5];
    const float* b_out  = (const float*)d_in[6];
    const float* A_log  = (const float*)d_in[7];
    float* y = (float*)d_out;

    const size_t SZ = (size_t)B_ * N_ * L_;   // 524288 floats
    float* ws    = (float*)d_ws;
    float* abar  = ws;
    float* bbar  = ws + SZ;
    float* cmean = ws + 2 * SZ;
    float* ch    = ws + 3 * SZ;               // (B,L,N)

    // K1: conv + input projection + pointwise -> abar/bbar/cmean (B,N,L)
    ssm_conv_proj_kernel<<<dim3((B_ * L_) / 16), dim3(160), 0, stream>>>(
        x, conv_w, conv_b, Win, b_in, A_log, abar, bbar, cmean);

    // K2: recurrence scan -> ch = C*h (B,L,N)
    ssm_scan_kernel<<<dim3(B_ * N_), dim3(32), 0, stream>>>(abar, bbar, cmean, ch);

    // K3: output projection + bias -> y
    ssm_out_kernel<<<dim3((B_ * L_) / 16, D_ / (16 * 8)), dim3(256), 0, stream>>>(
        ch, Wout, b_out, y);
}